// NegLoss_15719580304254
// MI455X (gfx1250) — compile-verified
//
#include <hip/hip_runtime.h>
#include <hip/hip_bf16.h>
#include <stdint.h>

#define EPS_F 1e-12f

typedef float v2f __attribute__((ext_vector_type(2)));
typedef float v8f __attribute__((ext_vector_type(8)));

// loss term for a single logit:  l^2 * (-max(log1p(-l), -100)) * label_weight
__device__ __forceinline__ float loss_term(float l, float lw) {
    float b = fmaxf(log1pf(-l), -100.0f);
    return l * l * (-b) * lw;
}

// Convert a 64-byte bool row (loaded as 4 x uint4) into a 64-bit mask.
__device__ __forceinline__ uint64_t mask_bits_row(const uint4* __restrict__ mrow) {
    uint64_t m = 0;
#pragma unroll
    for (int i = 0; i < 4; ++i) {
        uint4 v = mrow[i];
        uint32_t a0 = v.x, a1 = v.y, a2 = v.z, a3 = v.w;
        int base = i * 16;
        if (a0 & 0x000000FFu) m |= 1ull << (base + 0);
        if (a0 & 0x0000FF00u) m |= 1ull << (base + 1);
        if (a0 & 0x00FF0000u) m |= 1ull << (base + 2);
        if (a0 & 0xFF000000u) m |= 1ull << (base + 3);
        if (a1 & 0x000000FFu) m |= 1ull << (base + 4);
        if (a1 & 0x0000FF00u) m |= 1ull << (base + 5);
        if (a1 & 0x00FF0000u) m |= 1ull << (base + 6);
        if (a1 & 0xFF000000u) m |= 1ull << (base + 7);
        if (a2 & 0x000000FFu) m |= 1ull << (base + 8);
        if (a2 & 0x0000FF00u) m |= 1ull << (base + 9);
        if (a2 & 0x00FF0000u) m |= 1ull << (base + 10);
        if (a2 & 0xFF000000u) m |= 1ull << (base + 11);
        if (a3 & 0x000000FFu) m |= 1ull << (base + 12);
        if (a3 & 0x0000FF00u) m |= 1ull << (base + 13);
        if (a3 & 0x00FF0000u) m |= 1ull << (base + 14);
        if (a3 & 0xFF000000u) m |= 1ull << (base + 15);
    }
    return m;
}

__global__ void init_kernel(unsigned* __restrict__ gmin, unsigned* __restrict__ gmax,
                            float* __restrict__ partials, int nb, int npad) {
    int t = threadIdx.x;
    if (t < 64) { gmin[t] = 0x7F800000u; gmax[t] = 0u; }  // +inf / 0 (w is always >= 1)
    for (int i = nb + t; i < npad; i += blockDim.x) partials[i] = 0.0f;
}

// Pass A: per-gt wmin/wmax over masked points only. w > 0 always, so the
// positive-float == uint ordering trick makes atomicMin/Max exact & order-independent.
__global__ void minmax_kernel(const float* __restrict__ ious,
                              const uint8_t* __restrict__ mask,
                              unsigned* __restrict__ gmin, unsigned* __restrict__ gmax,
                              int P) {
    __shared__ unsigned smin[64], smax[64];
    int tid = threadIdx.x;
    if (tid < 64) { smin[tid] = 0x7F800000u; smax[tid] = 0u; }
    __syncthreads();
    int p = blockIdx.x * blockDim.x + tid;
    if (p < P) {
        const uint4* mrow = (const uint4*)(mask + (size_t)p * 64);
        uint64_t m = mask_bits_row(mrow);
        while (m) {
            int g = __builtin_ctzll(m);
            m &= m - 1;
            float iou = ious[(size_t)p * 64 + g];
            float w = 1.0f / fmaxf(1.0f - iou, EPS_F);
            unsigned wb = __float_as_uint(w);
            atomicMin(&smin[g], wb);
            atomicMax(&smax[g], wb);
        }
    }
    __syncthreads();
    if (tid < 64) {
        atomicMin(&gmin[tid], smin[tid]);
        atomicMax(&gmax[tid], smax[tid]);
    }
}

// Pass B: base terms (W==1) + sparse corrections for overridden classes.
// Descending-g with an 80-bit done-set == last-scan-iteration-wins semantics.
__global__ void loss_kernel(const float* __restrict__ cls,
                            const float* __restrict__ obj,
                            const int* __restrict__ labels,
                            const float* __restrict__ ious,
                            const float* __restrict__ lwts,
                            const uint8_t* __restrict__ mask,
                            const unsigned* __restrict__ gmin,
                            const unsigned* __restrict__ gmax,
                            float* __restrict__ partials,
                            int P, int C) {
    __shared__ int   s_lbl[64];
    __shared__ float s_wmin[64], s_wmax[64];
    __shared__ float red[256];
    int tid = threadIdx.x;
    if (tid < 64) {
        s_lbl[tid]  = labels[tid];
        s_wmin[tid] = __uint_as_float(gmin[tid]);
        s_wmax[tid] = __uint_as_float(gmax[tid]);
    }
    __syncthreads();

    int p = blockIdx.x * blockDim.x + tid;
    float sum = 0.0f;
    if (p < P) {
        float o  = obj[p];
        float lw = lwts[p];
        const float4* crow = (const float4*)(cls + (size_t)p * C);
        int n4 = C >> 2;
        for (int i = 0; i < n4; ++i) {
            float4 c4 = crow[i];
            sum += loss_term(c4.x * o, lw);
            sum += loss_term(c4.y * o, lw);
            sum += loss_term(c4.z * o, lw);
            sum += loss_term(c4.w * o, lw);
        }
        const uint4* mrow = (const uint4*)(mask + (size_t)p * 64);
        uint64_t m = mask_bits_row(mrow);
        uint64_t done_lo = 0; uint32_t done_hi = 0;
        while (m) {
            int g = 63 - __builtin_clzll(m);   // highest g first (scan winner)
            m &= ~(1ull << g);
            int lbl = s_lbl[g];
            bool dup;
            if (lbl < 64) { uint64_t b = 1ull << lbl;        dup = (done_lo & b) != 0; done_lo |= b; }
            else          { uint32_t b = 1u  << (lbl - 64);  dup = (done_hi & b) != 0; done_hi |= b; }
            if (dup) continue;
            float iou  = ious[(size_t)p * 64 + g];
            float w    = 1.0f / fmaxf(1.0f - iou, EPS_F);
            float norm = (w - s_wmin[g] + EPS_F) / (s_wmax[g] - s_wmin[g] + EPS_F);
            float upd  = 1.0f - norm;
            float lb   = cls[(size_t)p * C + lbl] * o;  // L2/L0 hit: row just streamed
            float ln   = lb * upd;
            sum += loss_term(ln, lw) - loss_term(lb, lw);
        }
    }
    red[tid] = sum;
    __syncthreads();
    for (int s = 128; s > 0; s >>= 1) {
        if (tid < s) red[tid] += red[tid + s];
        __syncthreads();
    }
    if (tid == 0) partials[blockIdx.x] = red[0];
}

// Pass C: deterministic final reduction via V_WMMA_F32_16X16X4_F32.
// 64 partials -> 16x4 A-matrix, B = ones(4x16); D[m,n] accumulates row sums
// (all columns equal). Column N=0 lives in lane 0 (M=0..7) and lane 16 (M=8..15).
__global__ void wmma_reduce_kernel(const float* __restrict__ partials,
                                   const int* __restrict__ avg_factor,
                                   float* __restrict__ out, int npad) {
    int lane = threadIdx.x;  // 32 threads == 1 wave, EXEC all ones for WMMA
    v8f acc = {};
    v2f bones; bones[0] = 1.0f; bones[1] = 1.0f;
    for (int base = 0; base < npad; base += 64) {
        v2f a;
        a[0] = partials[base + lane * 2 + 0];
        a[1] = partials[base + lane * 2 + 1];
        acc = __builtin_amdgcn_wmma_f32_16x16x4_f32(
            /*neg_a=*/false, a, /*neg_b=*/false, bones,
            /*c_mod=*/(short)0, acc, /*reuse_a=*/false, /*reuse_b=*/false);
    }
    float s = acc[0] + acc[1] + acc[2] + acc[3] + acc[4] + acc[5] + acc[6] + acc[7];
    float t0  = __shfl(s, 0, 32);   // M=0..7  column 0
    float t16 = __shfl(s, 16, 32);  // M=8..15 column 0
    if (lane == 0) out[0] = (t0 + t16) / (float)avg_factor[0];  // LOSS_WEIGHT == 1
}

extern "C" void kernel_launch(void* const* d_in, const int* in_sizes, int n_in,
                              void* d_out, int out_size, void* d_ws, size_t ws_size,
                              hipStream_t stream) {
    const float*   cls    = (const float*)d_in[0];     // (P, C)
    const float*   obj    = (const float*)d_in[1];     // (P, 1)
    const int*     labels = (const int*)d_in[2];       // (G,)
    const float*   ious   = (const float*)d_in[3];     // (P, G)
    const float*   lwts   = (const float*)d_in[4];     // (P, 1)
    const uint8_t* mask   = (const uint8_t*)d_in[5];   // (P, G) bool
    const int*     avgf   = (const int*)d_in[6];       // scalar

    int P = in_sizes[1];           // 262144
    int C = in_sizes[0] / P;       // 80

    unsigned* gmin     = (unsigned*)d_ws;
    unsigned* gmax     = gmin + 64;
    float*    partials = (float*)d_ws + 128;

    int nb   = (P + 255) / 256;          // 1024 blocks
    int npad = (nb + 63) & ~63;          // pad to WMMA chunk of 64

    init_kernel<<<1, 64, 0, stream>>>(gmin, gmax, partials, nb, npad);
    minmax_kernel<<<nb, 256, 0, stream>>>(ious, mask, gmin, gmax, P);
    loss_kernel<<<nb, 256, 0, stream>>>(cls, obj, labels, ious, lwts, mask,
                                        gmin, gmax, partials, P, C);
    wmma_reduce_kernel<<<1, 32, 0, stream>>>(partials, avgf, (float*)d_out, npad);
}